// Triangle_39719857553609
// MI455X (gfx1250) — compile-verified
//
#include <hip/hip_runtime.h>

// Problem constants (from the reference).
#define N_ATOMS    64
#define NC2        2016          // 64*63/2
#define MAT_ELEMS  (N_ATOMS * N_ATOMS)   // 4096
#define LDS_FLOATS 2048          // 2016 + slack (diag dummy index reaches 2016)

// ---- CDNA5 async global->LDS helpers (gfx1250) --------------------------

#if defined(__has_builtin)
#  if __has_builtin(__builtin_amdgcn_global_load_async_to_lds_b128)
#    define HAVE_ASYNC_LDS_BUILTIN 1
#  endif
#  if __has_builtin(__builtin_amdgcn_s_wait_asynccnt)
#    define HAVE_WAIT_ASYNC_BUILTIN 1
#  endif
#endif

// Builtin signature (confirmed by clang diagnostic):
//   void __builtin_amdgcn_global_load_async_to_lds_b128(
//       v4i __device__* src_global, v4i __shared__* dst_lds,
//       int imm_offset, int imm_cpol)
typedef int v4i __attribute__((vector_size(16)));
typedef __attribute__((address_space(1))) v4i as1_v4i;  // global
typedef __attribute__((address_space(3))) v4i as3_v4i;  // LDS

// Copy 16 bytes global -> LDS through the async (ASYNCcnt) path.
// Both addresses must be 16B aligned.
__device__ __forceinline__ void async_copy_b128(const float* __restrict__ g,
                                                float* __restrict__ l) {
#if defined(HAVE_ASYNC_LDS_BUILTIN)
  __builtin_amdgcn_global_load_async_to_lds_b128((as1_v4i*)g, (as3_v4i*)l,
                                                 /*offset=*/0, /*cpol=*/0);
#else
  // ISA 15.18.3 opcode 98: global_load_async_to_lds_b128 vdst(lds), vaddr, off
  unsigned lds_off = (unsigned)(unsigned long long)(as3_v4i*)l;
  asm volatile("global_load_async_to_lds_b128 %0, %1, off"
               :
               : "v"(lds_off), "v"(g)
               : "memory");
#endif
}

__device__ __forceinline__ void wait_async_zero() {
#if defined(HAVE_WAIT_ASYNC_BUILTIN)
  __builtin_amdgcn_s_wait_asynccnt(0);
#else
  asm volatile("s_wait_asynccnt 0" ::: "memory");
#endif
}

// ---- Kernel: one block per batch row ------------------------------------
// Stage 2016 floats into LDS via the async-DMA path, then each thread
// gathers 16 contiguous outputs of the symmetrized 64x64 matrix
// (branchless) and stores four float4s.

__global__ __launch_bounds__(256) void triangle_symmetrize_kernel(
    const float* __restrict__ in, float* __restrict__ out) {
  __shared__ __align__(16) float lds[LDS_FLOATS];

  const int tid = threadIdx.x;
  const int b   = blockIdx.x;

  const float* row_in = in + (size_t)b * NC2;  // 8064 B, 16B aligned (504*16)

  // 504 chunks of 16B, both passes unconditional (no exec-mask branch):
  // pass 1 = chunks [0,256); pass 2 = chunks [256,504), lanes past the end
  // redundantly re-copy chunk 503 (same src -> same dst, benign).
  async_copy_b128(row_in + tid * 4, &lds[tid * 4]);
  const int c2 = (256 + tid < 503) ? (256 + tid) : 503;
  async_copy_b128(row_in + c2 * 4, &lds[c2 * 4]);

  wait_async_zero();   // this wave's async LDS writes complete
  __syncthreads();     // make them visible to all waves in the workgroup

  // Output assignment: thread t -> row i = t/4, cols j0..j0+15, j0 = (t%4)*16.
  const int i  = tid >> 2;
  const int j0 = (tid & 3) << 4;

  float vals[16];
#pragma unroll
  for (int k = 0; k < 16; ++k) {
    const int j  = j0 + k;
    const int hi = (i > j) ? i : j;          // v_max_i32
    const int lo = i + j - hi;               // = min(i,j)
    const float v = lds[((hi * (hi - 1)) >> 1) + lo];  // symmetric gather
    vals[k] = (i == j) ? 0.0f : v;           // v_cndmask: diagonal -> 0
  }

  // 4x float4 = 64 contiguous bytes per thread; wave32 -> 2KB coalesced burst.
  float4* outp = (float4*)(out + (size_t)b * MAT_ELEMS +
                           (size_t)i * N_ATOMS + (size_t)j0);
#pragma unroll
  for (int q = 0; q < 4; ++q) {
    outp[q] = make_float4(vals[4 * q + 0], vals[4 * q + 1],
                          vals[4 * q + 2], vals[4 * q + 3]);
  }
}

// ---- Entry point ---------------------------------------------------------

extern "C" void kernel_launch(void* const* d_in, const int* in_sizes, int n_in,
                              void* d_out, int out_size, void* d_ws,
                              size_t ws_size, hipStream_t stream) {
  const float* in  = (const float*)d_in[0];
  float*       out = (float*)d_out;

  const int batch = in_sizes[0] / NC2;  // 32768 for the reference shapes

  triangle_symmetrize_kernel<<<batch, 256, 0, stream>>>(in, out);
}